// BipolarMorphologicalConv2D_7155415515403
// MI455X (gfx1250) — compile-verified
//
#include <hip/hip_runtime.h>
#include <math.h>
#include <stdint.h>

// Bipolar morphological conv2d (max-plus semiring): WMMA is inapplicable
// (non-linear reduction over (max,+) — not expressible as D = A*B + C).
// CDNA5 features used instead:
//  - global_load_async_to_lds_b128 (ASYNCcnt DMA) to stage x-rows + both
//    kernels into LDS once per workgroup
//  - s_wait_asynccnt 0 for completion
//  - wave32 layout: lanes = cout -> lp LDS reads are wave-uniform broadcast
//    ds_load_b128s, kernel LDS reads and global stores are fully coalesced.

#define FH    3
#define FW    3
#define CIN   32
#define COUT  64
#define HIN   32
#define WIN   32
#define HO    30
#define WO    30
#define NB    8

// LDS layout (byte offsets inside dynamic smem):
static constexpr int LDS_XRAW = 0;                    // 3*32*32 f32 = 12288 B
static constexpr int LDS_LX1  = 12288;                // 12288 B
static constexpr int LDS_LX2  = 24576;                // 12288 B
static constexpr int LDS_K1   = 36864;                // 288*64 f32 = 73728 B
static constexpr int LDS_K2   = 36864 + 73728;        // 73728 B
static constexpr int LDS_TOTAL = 36864 + 2 * 73728;   // 184320 B (< 320 KB WGP)

__device__ __forceinline__ void async_copy_b128(uint32_t lds_byte_addr,
                                                const void* gptr) {
  // Per-lane: copy 16B global -> LDS. VDST VGPR holds LDS byte address,
  // VADDR pair holds the 64-bit global address (GV mode, saddr = off).
  asm volatile("global_load_async_to_lds_b128 %0, %1, off"
               :
               : "v"(lds_byte_addr), "v"((uint64_t)(uintptr_t)gptr)
               : "memory");
}

__device__ __forceinline__ void wait_async_zero() {
#if __has_builtin(__builtin_amdgcn_s_wait_asynccnt)
  __builtin_amdgcn_s_wait_asynccnt(0);
#else
  asm volatile("s_wait_asynccnt 0x0" ::: "memory");
#endif
}

__global__ __launch_bounds__(256, 1) void bipolar_morph_conv_kernel(
    const float* __restrict__ x,     // [8,32,32,32]
    const float* __restrict__ k1,    // [3,3,32,64]
    const float* __restrict__ k2,    // [3,3,32,64]
    const float* __restrict__ bias,  // [64]
    float* __restrict__ out) {       // [8,30,30,64]
  extern __shared__ char smem[];
  // Low 32 bits of the generic LDS address == byte offset in wave LDS space.
  const uint32_t lds_base = (uint32_t)(uintptr_t)(void*)smem;

  const int t  = threadIdx.x;
  const int bx = blockIdx.x;       // 0 .. 239
  const int n  = bx / HO;
  const int ho = bx % HO;

  // ---- Phase 1: async DMA staging into LDS --------------------------------
  // Rows ho..ho+2 of x[n] are contiguous: 3*32*32 floats = 768 x 16B chunks.
  const char* xrow = (const char*)(x + (size_t)(n * HIN + ho) * WIN * CIN);
#pragma unroll
  for (int j = 0; j < 3; ++j)
    async_copy_b128(lds_base + LDS_XRAW + (t + 256 * j) * 16,
                    xrow + (t + 256 * j) * 16);
  // Kernels: 288*64 floats = 4608 x 16B chunks each (18 per thread).
#pragma unroll 2
  for (int j = 0; j < 18; ++j) {
    async_copy_b128(lds_base + LDS_K1 + (t + 256 * j) * 16,
                    (const char*)k1 + (t + 256 * j) * 16);
    async_copy_b128(lds_base + LDS_K2 + (t + 256 * j) * 16,
                    (const char*)k2 + (t + 256 * j) * 16);
  }
  wait_async_zero();
  __syncthreads();

  const float* xraw = (const float*)(smem + LDS_XRAW);
  float* lx1 = (float*)(smem + LDS_LX1);
  float* lx2 = (float*)(smem + LDS_LX2);
  const float* k1s = (const float*)(smem + LDS_K1);
  const float* k2s = (const float*)(smem + LDS_K2);

  // ---- Phase 2: log(max(+/-x, 0.1)) once, shared by all 64 couts ----------
  for (int i = t; i < 3 * WIN * CIN; i += 256) {
    float v = xraw[i];
    lx1[i] = __logf(fmaxf(v, 0.1f));
    lx2[i] = __logf(fmaxf(-v, 0.1f));
  }
  __syncthreads();

  // ---- Phase 3: max-plus reduction ----------------------------------------
  // Thread -> (cout, wo-group). Lanes of a wave share the wo-group, so lp
  // loads are wave-uniform LDS broadcasts; k loads are lane-consecutive.
  const int c  = t & 63;
  const int wg = t >> 6;  // 0..3 ; wo = wg + 4*w, w = 0..7

  float m11[8], m12[8], m21[8], m22[8];
#pragma unroll
  for (int w = 0; w < 8; ++w) {
    m11[w] = -INFINITY; m12[w] = -INFINITY;
    m21[w] = -INFINITY; m22[w] = -INFINITY;
  }

  for (int fh = 0; fh < FH; ++fh) {
    for (int fw = 0; fw < FW; ++fw) {
      const int pbase = (fh * FW + fw) * CIN;      // p = pbase + ci
      const int rowb  = fh * WIN * CIN;            // lx row base
      int colb[8];
#pragma unroll
      for (int w = 0; w < 8; ++w) {
        int wo = wg + 4 * w;
        if (wo > WO - 1) wo = WO - 1;              // clamp (store is guarded)
        colb[w] = rowb + (wo + fw) * CIN;          // multiple of 32 floats
      }
      // ci unrolled by 4: lp loads become 16B ds_load_b128 broadcasts.
      for (int ci = 0; ci < CIN; ci += 4) {
        float kv1[4], kv2[4];
#pragma unroll
        for (int q = 0; q < 4; ++q) {
          kv1[q] = k1s[(pbase + ci + q) * COUT + c];
          kv2[q] = k2s[(pbase + ci + q) * COUT + c];
        }
#pragma unroll
        for (int w = 0; w < 8; ++w) {
          const float4 a1 = *(const float4*)(lx1 + colb[w] + ci);
          const float4 a2 = *(const float4*)(lx2 + colb[w] + ci);
          m11[w] = fmaxf(m11[w], a1.x + kv1[0]);
          m11[w] = fmaxf(m11[w], a1.y + kv1[1]);
          m11[w] = fmaxf(m11[w], a1.z + kv1[2]);
          m11[w] = fmaxf(m11[w], a1.w + kv1[3]);
          m12[w] = fmaxf(m12[w], a1.x + kv2[0]);
          m12[w] = fmaxf(m12[w], a1.y + kv2[1]);
          m12[w] = fmaxf(m12[w], a1.z + kv2[2]);
          m12[w] = fmaxf(m12[w], a1.w + kv2[3]);
          m21[w] = fmaxf(m21[w], a2.x + kv1[0]);
          m21[w] = fmaxf(m21[w], a2.y + kv1[1]);
          m21[w] = fmaxf(m21[w], a2.z + kv1[2]);
          m21[w] = fmaxf(m21[w], a2.w + kv1[3]);
          m22[w] = fmaxf(m22[w], a2.x + kv2[0]);
          m22[w] = fmaxf(m22[w], a2.y + kv2[1]);
          m22[w] = fmaxf(m22[w], a2.z + kv2[2]);
          m22[w] = fmaxf(m22[w], a2.w + kv2[3]);
        }
      }
    }
  }

  // ---- Phase 4: exp-combine + bias, coalesced store -----------------------
  const float b = bias[c];
  float* orow = out + ((size_t)(n * HO + ho) * WO) * COUT + c;
#pragma unroll
  for (int w = 0; w < 8; ++w) {
    const int wo = wg + 4 * w;
    if (wo < WO) {
      const float y = __expf(m11[w]) - __expf(m12[w])
                    - __expf(m21[w]) + __expf(m22[w]) + b;
      orow[wo * COUT] = y;
    }
  }
}

extern "C" void kernel_launch(void* const* d_in, const int* in_sizes, int n_in,
                              void* d_out, int out_size, void* d_ws, size_t ws_size,
                              hipStream_t stream) {
  (void)in_sizes; (void)n_in; (void)out_size; (void)d_ws; (void)ws_size;
  const float* x    = (const float*)d_in[0];
  const float* k1   = (const float*)d_in[1];
  const float* k2   = (const float*)d_in[2];
  const float* bias = (const float*)d_in[3];
  float* out        = (float*)d_out;

  dim3 grid(NB * HO);   // 240 workgroups: one output row each
  dim3 block(256);      // 8 wave32 waves
  hipLaunchKernelGGL(bipolar_morph_conv_kernel, grid, block, LDS_TOTAL, stream,
                     x, k1, k2, bias, out);
}